// TopkRouter_50946902065584
// MI455X (gfx1250) — compile-verified
//
#include <hip/hip_runtime.h>

typedef __attribute__((ext_vector_type(2))) float v2f;
typedef __attribute__((ext_vector_type(4))) float v4f;
typedef __attribute__((ext_vector_type(8))) float v8f;

#define TOKENS     16384            // B*S
#define DDIM       2048
#define NEXP       64
#define KC         64               // K-chunk staged in LDS per pipeline stage
#define KGROUP     1024             // split-K: each wave-group handles half of D
#define NCHUNK     (KGROUP / KC)    // 16 pipeline stages
#define LDS_PITCH  68               // 64 + 4 pad -> conflict-free b128 reads
#define CHUNK_F    (NEXP * LDS_PITCH)   // 4352 floats per staged chunk
#define TILES_PB   4                // token tiles per block (8 waves = 4 tiles x 2 K-groups)

__device__ __forceinline__ v8f wmma4(v2f a, v2f b, v8f c) {
    return __builtin_amdgcn_wmma_f32_16x16x4_f32(
        false, a, false, b, (short)0, c, false, false);
}

// Direct global->LDS DMA: 8 x b128 per thread; the same instruction offset
// advances BOTH the LDS destination and global source (ISA 10.x async pseudocode).
// Tracked with ASYNCcnt.
__device__ __forceinline__ void async_copy_b128x8(unsigned lds_off, const float* gptr) {
    unsigned long long ga = (unsigned long long)(uintptr_t)gptr;
    asm volatile(
        "global_load_async_to_lds_b128 %0, %1, off\n\t"
        "global_load_async_to_lds_b128 %0, %1, off offset:16\n\t"
        "global_load_async_to_lds_b128 %0, %1, off offset:32\n\t"
        "global_load_async_to_lds_b128 %0, %1, off offset:48\n\t"
        "global_load_async_to_lds_b128 %0, %1, off offset:64\n\t"
        "global_load_async_to_lds_b128 %0, %1, off offset:80\n\t"
        "global_load_async_to_lds_b128 %0, %1, off offset:96\n\t"
        "global_load_async_to_lds_b128 %0, %1, off offset:112"
        :: "v"(lds_off), "v"(ga) : "memory");
}

__device__ __forceinline__ void wait_async0() {
    asm volatile("s_wait_asynccnt 0x0" ::: "memory");
}

__device__ __forceinline__ void lds_store_v8(float* p, v8f v) {
    float4 lo = {v[0], v[1], v[2], v[3]};
    float4 hi = {v[4], v[5], v[6], v[7]};
    reinterpret_cast<float4*>(p)[0] = lo;
    reinterpret_cast<float4*>(p)[1] = hi;
}

__device__ __forceinline__ v8f lds_add_v8(v8f v, const float* p) {
    float4 lo = reinterpret_cast<const float4*>(p)[0];
    float4 hi = reinterpret_cast<const float4*>(p)[1];
    v[0] += lo.x; v[1] += lo.y; v[2] += lo.z; v[3] += lo.w;
    v[4] += hi.x; v[5] += hi.y; v[6] += hi.z; v[7] += hi.w;
    return v;
}

__global__ __launch_bounds__(256) void router_wmma_kernel(
    const float* __restrict__ x,      // [TOKENS, DDIM]
    const float* __restrict__ W,      // [NEXP, DDIM]
    const float* __restrict__ bias,   // [NEXP]
    float* __restrict__ mask_out,     // [TOKENS, NEXP]  (0.0 / 1.0)
    float* __restrict__ prob_out)     // [TOKENS, NEXP]
{
    // [buf(2)][kgroup(2)][CHUNK_F] = 4*4352 floats = 69,632 B
    __shared__ float lds[4 * CHUNK_F];

    const int tid   = threadIdx.x;
    const int wave  = tid >> 5;
    const int lane  = tid & 31;
    const int n     = lane & 15;        // expert-within-tile / A row
    const int khalf = lane >> 4;        // 0 -> K slots 0..1 ; 1 -> K slots 2..3
    const int koff  = khalf * 4;        // contiguous 4-float chunk per lane
    const int tile  = wave & 3;         // token tile within block
    const int kg    = wave >> 2;        // split-K group (0 or 1)
    const int tokenBase = (blockIdx.x * TILES_PB + tile) * 16;

    v8f acc0 = {}, acc1 = {}, acc2 = {}, acc3 = {};

    // A: lane L holds row M = L%16; this wave-group's K window
    const float* xrow = x + (size_t)(tokenBase + n) * DDIM + kg * KGROUP + koff;

    // W staging roles: 128 threads per K-group stage that group's 64x64 chunk
    const int gt   = tid & 127;
    const int sgrp = tid >> 7;              // which K-group this thread stages
    const int srow = gt >> 1;               // expert row 0..63
    const int scol = (gt & 1) * 32;         // 32-float half-row
    const float* wsrc = W + (size_t)srow * DDIM + sgrp * KGROUP + scol;
    // flat-LDS low 32 bits == wave-relative LDS byte offset (ISA 10.2 aperture map)
    const unsigned sdst_off =
        (unsigned)(uintptr_t)&lds[sgrp * CHUNK_F + srow * LDS_PITCH + scol];
    const unsigned buf_stride_b = (unsigned)(2 * CHUNK_F * sizeof(float));

    // ---- software pipeline: async global->LDS DMA, 1 barrier/chunk --------
    async_copy_b128x8(sdst_off, wsrc);      // chunk 0 -> buffer 0
    wait_async0();
    __syncthreads();

    const float* bgrp = &lds[kg * CHUNK_F];
    // A pipelined one K-step ahead
    v4f a4 = __builtin_nontemporal_load(reinterpret_cast<const v4f*>(xrow));

    for (int ci = 0; ci < NCHUNK; ++ci) {
        if (ci + 1 < NCHUNK)                // DMA next chunk while computing
            async_copy_b128x8(sdst_off + ((ci + 1) & 1) * buf_stride_b,
                              wsrc + (ci + 1) * KC);

        const float* bp = bgrp + (ci & 1) * (2 * CHUNK_F) + koff;

        #pragma unroll
        for (int k8 = 0; k8 < KC; k8 += 8) {
            const int gk = ci * KC + k8;
            // issue next A load (uniform select; last step harmlessly reloads k=0)
            const float* nap = xrow + ((gk + 8 < KGROUP) ? gk + 8 : 0);
            v4f a_next = __builtin_nontemporal_load(
                reinterpret_cast<const v4f*>(nap));

            v2f a1 = {a4.x, a4.y};          // K-perm {k,k+1,k+4,k+5}
            v2f a2 = {a4.z, a4.w};          // K-perm {k+2,k+3,k+6,k+7}

            // 4 B tiles into distinct regs -> graduated dscnt waits
            v4f b0 = *reinterpret_cast<const v4f*>(bp + (n     ) * LDS_PITCH + k8);
            v4f b1 = *reinterpret_cast<const v4f*>(bp + (16 + n) * LDS_PITCH + k8);
            v4f b2 = *reinterpret_cast<const v4f*>(bp + (32 + n) * LDS_PITCH + k8);
            v4f b3 = *reinterpret_cast<const v4f*>(bp + (48 + n) * LDS_PITCH + k8);

            // interleaved: distance-4 on each accumulator RAW
            acc0 = wmma4(a1, (v2f){b0.x, b0.y}, acc0);
            acc1 = wmma4(a1, (v2f){b1.x, b1.y}, acc1);
            acc2 = wmma4(a1, (v2f){b2.x, b2.y}, acc2);
            acc3 = wmma4(a1, (v2f){b3.x, b3.y}, acc3);
            acc0 = wmma4(a2, (v2f){b0.z, b0.w}, acc0);
            acc1 = wmma4(a2, (v2f){b1.z, b1.w}, acc1);
            acc2 = wmma4(a2, (v2f){b2.z, b2.w}, acc2);
            acc3 = wmma4(a2, (v2f){b3.z, b3.w}, acc3);

            a4 = a_next;
        }

        if (ci + 1 < NCHUNK) wait_async0(); // DMA must land before barrier
        __syncthreads();
    }

    // ---- split-K reduction through LDS (36-float lane stride, no conflicts) ----
    if (kg == 1) {
        float* dst = &lds[(wave - 4) * (32 * 36) + lane * 36];
        lds_store_v8(dst +  0, acc0);
        lds_store_v8(dst +  8, acc1);
        lds_store_v8(dst + 16, acc2);
        lds_store_v8(dst + 24, acc3);
    }
    __syncthreads();
    if (kg == 1) return;                    // all WMMAs done; divergence is safe now

    {
        const float* src = &lds[wave * (32 * 36) + lane * 36];
        acc0 = lds_add_v8(acc0, src +  0);
        acc1 = lds_add_v8(acc1, src +  8);
        acc2 = lds_add_v8(acc2, src + 16);
        acc3 = lds_add_v8(acc3, src + 24);
    }

    // ---- Epilogue: bias + softmax(E=64) + top-2 mask ----------------------
    const float b0 = bias[n], b1 = bias[16 + n], b2 = bias[32 + n], b3 = bias[48 + n];
    const float NEG_INF = -3.402823466e38f;

    #pragma unroll
    for (int j = 0; j < 8; ++j) {
        // token for this (j, lane-half): M = j + 8*khalf
        float lv0 = acc0[j] + b0;
        float lv1 = acc1[j] + b1;
        float lv2 = acc2[j] + b2;
        float lv3 = acc3[j] + b3;

        float mx = fmaxf(fmaxf(lv0, lv1), fmaxf(lv2, lv3));
        #pragma unroll
        for (int off = 1; off < 16; off <<= 1)
            mx = fmaxf(mx, __shfl_xor(mx, off, 32));

        float e0 = __expf(lv0 - mx);
        float e1 = __expf(lv1 - mx);
        float e2 = __expf(lv2 - mx);
        float e3 = __expf(lv3 - mx);
        float s = e0 + e1 + e2 + e3;
        #pragma unroll
        for (int off = 1; off < 16; off <<= 1)
            s += __shfl_xor(s, off, 32);
        const float inv = 1.0f / s;

        // top-1 (lowest index wins ties, matching lax.top_k)
        float m1 = lv0; int i1 = n;
        if (lv1 > m1) { m1 = lv1; i1 = 16 + n; }
        if (lv2 > m1) { m1 = lv2; i1 = 32 + n; }
        if (lv3 > m1) { m1 = lv3; i1 = 48 + n; }
        #pragma unroll
        for (int off = 1; off < 16; off <<= 1) {
            float ov = __shfl_xor(m1, off, 32);
            int   oi = __shfl_xor(i1, off, 32);
            if (ov > m1 || (ov == m1 && oi < i1)) { m1 = ov; i1 = oi; }
        }

        // top-2: max over the rest
        float m2 = NEG_INF; int i2 = 0x7fffffff;
        {
            float c0 = (i1 == n)      ? NEG_INF : lv0;
            float c1 = (i1 == 16 + n) ? NEG_INF : lv1;
            float c2 = (i1 == 32 + n) ? NEG_INF : lv2;
            float c3 = (i1 == 48 + n) ? NEG_INF : lv3;
            if (c0 > m2 || (c0 == m2 && n      < i2)) { m2 = c0; i2 = n; }
            if (c1 > m2 || (c1 == m2 && 16 + n < i2)) { m2 = c1; i2 = 16 + n; }
            if (c2 > m2 || (c2 == m2 && 32 + n < i2)) { m2 = c2; i2 = 32 + n; }
            if (c3 > m2 || (c3 == m2 && 48 + n < i2)) { m2 = c3; i2 = 48 + n; }
        }
        #pragma unroll
        for (int off = 1; off < 16; off <<= 1) {
            float ov = __shfl_xor(m2, off, 32);
            int   oi = __shfl_xor(i2, off, 32);
            if (ov > m2 || (ov == m2 && oi < i2)) { m2 = ov; i2 = oi; }
        }

        const int token = tokenBase + j + khalf * 8;
        float* mrow = mask_out + (size_t)token * NEXP;
        float* prow = prob_out + (size_t)token * NEXP;
        // write-once outputs -> non-temporal stores
        __builtin_nontemporal_store((n      == i1 || n      == i2) ? 1.0f : 0.0f, mrow + n);
        __builtin_nontemporal_store((16 + n == i1 || 16 + n == i2) ? 1.0f : 0.0f, mrow + 16 + n);
        __builtin_nontemporal_store((32 + n == i1 || 32 + n == i2) ? 1.0f : 0.0f, mrow + 32 + n);
        __builtin_nontemporal_store((48 + n == i1 || 48 + n == i2) ? 1.0f : 0.0f, mrow + 48 + n);
        __builtin_nontemporal_store(e0 * inv, prow + n);
        __builtin_nontemporal_store(e1 * inv, prow + 16 + n);
        __builtin_nontemporal_store(e2 * inv, prow + 32 + n);
        __builtin_nontemporal_store(e3 * inv, prow + 48 + n);
    }
}

extern "C" void kernel_launch(void* const* d_in, const int* in_sizes, int n_in,
                              void* d_out, int out_size, void* d_ws, size_t ws_size,
                              hipStream_t stream) {
    (void)in_sizes; (void)n_in; (void)out_size; (void)d_ws; (void)ws_size;
    const float* x    = (const float*)d_in[0];   // [4,4096,2048] -> [16384,2048]
    const float* W    = (const float*)d_in[1];   // [64,2048]
    const float* bias = (const float*)d_in[2];   // [64]
    // d_in[3] is k (==2), hardcoded in the kernel's top-2 logic.

    float* mask_out = (float*)d_out;                             // [16384,64]
    float* prob_out = (float*)d_out + (size_t)TOKENS * NEXP;     // [16384,64]

    dim3 grid(TOKENS / (16 * TILES_PB));   // 256 blocks of 256 threads (8 waves)
    router_wmma_kernel<<<grid, dim3(256), 0, stream>>>(x, W, bias, mask_out, prob_out);
}